// ChainCRF_11519102288033
// MI455X (gfx1250) — compile-verified
//
#include <hip/hip_runtime.h>
#include <math.h>

typedef float v2f __attribute__((ext_vector_type(2)));
typedef float v8f __attribute__((ext_vector_type(8)));

#define B_  128
#define S_  256
#define C_  64
#define LDS_STRIDE 68   // 16 rows, stride 68 floats -> conflict-free per-row reads

// ---------------------------------------------------------------------------
// Forward (free-energy) kernel: 8 blocks x 128 threads (4 waves).
// Block handles batches [b0, b0+16). Wave w owns output columns [16w, 16w+16).
// logsumexp recursion done as f32 WMMA GEMM against precomputed exp(U).
// ---------------------------------------------------------------------------
__global__ __launch_bounds__(128) void crf_forward_kernel(
    const float* __restrict__ em, const float* __restrict__ mask,
    const float* __restrict__ U, float* __restrict__ free_out)
{
  __shared__ float abuf[2][16 * LDS_STRIDE];

  const int tid  = threadIdx.x;
  const int lane = tid & 31;
  const int wid  = tid >> 5;
  const int half = lane >> 4;     // lane group (0: lanes 0-15, 1: lanes 16-31)
  const int row  = lane & 15;     // A-matrix row this lane feeds
  const int col  = row + 16 * wid;// C/D column this lane owns
  const int b0   = blockIdx.x * 16;

  // B operand tiles: exp(U)[K][col] for 16 K-slices of 4.
  // ISA layout for 4x16 B (2 VGPRs): lane L, vgpr v -> K = v + 2*(L/16).
  v2f Breg[16];
#pragma unroll
  for (int kk = 0; kk < 16; ++kk) {
    const int k0 = 4 * kk + 2 * half;
    Breg[kk].x = __expf(U[(k0 + 0) * C_ + col]);
    Breg[kk].y = __expf(U[(k0 + 1) * C_ + col]);
  }

  // alpha0 = emissions[:, 0, :]
  for (int idx = tid; idx < 16 * C_; idx += 128) {
    const int r = idx >> 6, c = idx & 63;
    abuf[0][r * LDS_STRIDE + c] = em[(size_t)(b0 + r) * S_ * C_ + c];
  }
  __syncthreads();

  for (int t = 1; t < S_; ++t) {
    const float* src = abuf[(t - 1) & 1];
    float*       dst = abuf[t & 1];

    // Issue next-step emission/mask loads early to hide latency.
    float emv[8], mv[8];
#pragma unroll
    for (int v = 0; v < 8; ++v) {
      const int M = v + 8 * half;                 // C/D row of acc[v]
      emv[v] = em[((size_t)(b0 + M) * S_ + t) * C_ + col];
      mv[v]  = mask[(size_t)(b0 + M) * S_ + t];
    }
    if (t + 1 < S_)
      __builtin_prefetch(&em[((size_t)(b0 + row) * S_ + (t + 1)) * C_], 0, 0);

    // Row max for numerical stability (lane pairs redundantly per row).
    const float* srow = src + row * LDS_STRIDE;
    float mx = srow[0];
#pragma unroll
    for (int c = 1; c < C_; ++c) mx = fmaxf(mx, srow[c]);

    // A = exp(alpha - rowmax); accumulate 16 K-slices with f32 WMMA.
    v8f acc = {};
#pragma unroll
    for (int kk = 0; kk < 16; ++kk) {
      const int k0 = 4 * kk + 2 * half;           // K = v + 2*half + 4*kk
      v2f a;
      a.x = __expf(srow[k0 + 0] - mx);
      a.y = __expf(srow[k0 + 1] - mx);
      acc = __builtin_amdgcn_wmma_f32_16x16x4_f32(
          false, a, false, Breg[kk], (short)0, acc, false, false);
    }

    // alpha_new = log(acc) + rowmax(row M) + em*mask, write back to LDS.
#pragma unroll
    for (int v = 0; v < 8; ++v) {
      const int M   = v + 8 * half;
      const float mxv = __shfl(mx, M, 32);        // rowmax held by lane M
      dst[M * LDS_STRIDE + col] = __logf(acc[v]) + mxv + emv[v] * mv[v];
    }
    __syncthreads();   // publish dst before it becomes next step's src
  }

  // free_energy[b] = logsumexp(alpha_last); lanes 0..15 of wave 0.
  if (tid < 16) {
    const float* srow = abuf[(S_ - 1) & 1] + tid * LDS_STRIDE;
    float mx = srow[0];
    for (int c = 1; c < C_; ++c) mx = fmaxf(mx, srow[c]);
    float s = 0.f;
    for (int c = 0; c < C_; ++c) s += __expf(srow[c] - mx);
    free_out[b0 + tid] = mx + __logf(s);
  }
}

// ---------------------------------------------------------------------------
// Path energy: one block per batch, one thread per timestep.
// ---------------------------------------------------------------------------
__global__ __launch_bounds__(256) void path_energy_kernel(
    const float* __restrict__ em, const int* __restrict__ tags,
    const float* __restrict__ mask, const float* __restrict__ U,
    float* __restrict__ tag_out)
{
  __shared__ float red[8];
  const int b = blockIdx.x;
  const int t = threadIdx.x;

  const float m  = mask[(size_t)b * S_ + t];
  const int   tg = tags[(size_t)b * S_ + t];
  const int   tgm = (int)((float)tg * m);
  float e = em[((size_t)b * S_ + t) * C_ + tgm] * m;
  if (t >= 1) {
    const int p = tags[(size_t)b * S_ + t - 1];
    e += U[p * C_ + tg] * m;
  }
  for (int off = 16; off > 0; off >>= 1) e += __shfl_down(e, off, 32);
  if ((t & 31) == 0) red[t >> 5] = e;
  __syncthreads();
  if (t < 8) {
    float s = red[t];
    for (int off = 4; off > 0; off >>= 1) s += __shfl_down(s, off, 32);
    if (t == 0) tag_out[b] = s;
  }
}

// ---------------------------------------------------------------------------
// Finalize: mean(free_energy - tag_energy) over B=128.
// ---------------------------------------------------------------------------
__global__ __launch_bounds__(128) void finalize_kernel(
    const float* __restrict__ free_e, const float* __restrict__ tag_e,
    float* __restrict__ out)
{
  __shared__ float red[4];
  const int i = threadIdx.x;
  float d = free_e[i] - tag_e[i];
  for (int off = 16; off > 0; off >>= 1) d += __shfl_down(d, off, 32);
  if ((i & 31) == 0) red[i >> 5] = d;
  __syncthreads();
  if (i == 0) out[0] = (red[0] + red[1] + red[2] + red[3]) / (float)B_;
}

// ---------------------------------------------------------------------------
extern "C" void kernel_launch(void* const* d_in, const int* in_sizes, int n_in,
                              void* d_out, int out_size, void* d_ws, size_t ws_size,
                              hipStream_t stream) {
  const float* em   = (const float*)d_in[0];  // (B,S,C) f32
  const int*   tags = (const int*)d_in[1];    // (B,S)   i32
  const float* mask = (const float*)d_in[2];  // (B,S)   f32
  const float* U    = (const float*)d_in[3];  // (C,C)   f32
  float* out = (float*)d_out;

  float* free_e = (float*)d_ws;       // 128 floats
  float* tag_e  = free_e + B_;        // 128 floats

  crf_forward_kernel<<<B_ / 16, 128, 0, stream>>>(em, mask, U, free_e);
  path_energy_kernel<<<B_, S_, 0, stream>>>(em, tags, mask, U, tag_e);
  finalize_kernel<<<1, 128, 0, stream>>>(free_e, tag_e, out);
}